// AdaptivePatchEmbedding_24378234372403
// MI455X (gfx1250) — compile-verified
//
#include <hip/hip_runtime.h>
#include <hip/hip_bf16.h>
#include <math.h>

typedef __attribute__((ext_vector_type(2))) float v2f;
typedef __attribute__((ext_vector_type(8))) float v8f;

#define NBLK 43008            // B*C*R = 64*21*32
#define XLEN (NBLK * 32)      // total input samples
#define NROW (NBLK * 4)       // output rows (block,slot)
#define DM 512
#define LDA 97                // padded LDS stride (96 + 1) to avoid bank conflicts
#define OUT_MAIN (88080384)   // NROW * DM

// ---------------------------------------------------------------------------
// init: build zero-padded concatenated weight bank B[96][512] and PE[128][512]
// ---------------------------------------------------------------------------
__global__ __launch_bounds__(256) void init_kernel(
    const float* __restrict__ w_emb0, const float* __restrict__ w_emb1,
    const float* __restrict__ w_emb2, float* __restrict__ wsB,
    float* __restrict__ wsPE, float* __restrict__ d_out, int out_size) {
  int tid = blockIdx.x * 256 + threadIdx.x;
  const int NB  = 96 * DM;
  const int NPE = 128 * DM;
  if (tid == 0 && out_size > OUT_MAIN) d_out[OUT_MAIN] = 21.0f;  // scalar C
  if (tid < NB) {
    int q = tid / DM, d = tid % DM;
    float v = 0.0f;
    if (q < 8)                 v = w_emb0[d * 8  + q];
    else if (q >= 32 && q < 48) v = w_emb1[d * 16 + (q - 32)];
    else if (q >= 64)           v = w_emb2[d * 32 + (q - 64)];
    wsB[q * DM + d] = v;
  } else if (tid < NB + NPE) {
    int idx = tid - NB;
    int pos = idx / DM, d = idx % DM;
    int i = d >> 1;
    float freq = expf(-(float)(2 * i) * (logf(10000.0f) / (float)DM));
    float ang  = (float)pos * freq;
    wsPE[idx] = (d & 1) ? cosf(ang) : sinf(ang);
  }
}

// ---------------------------------------------------------------------------
// classifier: 32 -> 64 (relu) -> 3, argmax (first-max tie-break like jnp)
// ---------------------------------------------------------------------------
__global__ __launch_bounds__(256) void classify_kernel(
    const float* __restrict__ x, const float* __restrict__ w1,
    const float* __restrict__ b1, const float* __restrict__ w2,
    const float* __restrict__ b2, int* __restrict__ pred) {
  __shared__ float s_w1[64 * 32];
  __shared__ float s_b1[64];
  __shared__ float s_w2[3 * 64];
  __shared__ float s_b2[3];
  int t = threadIdx.x;
  for (int i = t; i < 64 * 32; i += 256) s_w1[i] = w1[i];
  if (t < 64)  s_b1[t] = b1[t];
  if (t < 192) s_w2[t] = w2[t];
  if (t < 3)   s_b2[t] = b2[t];
  __syncthreads();
  int blk = blockIdx.x * 256 + t;
  if (blk >= NBLK) return;
  float xv[32];
  const float* xp = x + (size_t)blk * 32;
#pragma unroll
  for (int p = 0; p < 32; ++p) xv[p] = xp[p];
  float l0 = s_b2[0], l1 = s_b2[1], l2 = s_b2[2];
  for (int h = 0; h < 64; ++h) {
    float a = s_b1[h];
    const float* wr = &s_w1[h * 32];
#pragma unroll
    for (int p = 0; p < 32; ++p) a = fmaf(wr[p], xv[p], a);
    a = fmaxf(a, 0.0f);
    l0 = fmaf(s_w2[h],       a, l0);
    l1 = fmaf(s_w2[64 + h],  a, l1);
    l2 = fmaf(s_w2[128 + h], a, l2);
  }
  int p = 0; float best = l0;
  if (l1 > best) { best = l1; p = 1; }
  if (l2 > best) { best = l2; p = 2; }
  pred[blk] = p;
}

// ---------------------------------------------------------------------------
// main GEMM: Out[64 rows x 512] per WG via v_wmma_f32_16x16x4_f32, K=96
// ---------------------------------------------------------------------------
__global__ __launch_bounds__(256) void embed_gemm_kernel(
    const float* __restrict__ x, const float* __restrict__ wsB,
    const float* __restrict__ wsPE, const int* __restrict__ pred,
    float* __restrict__ out) {
  __shared__ float sA[64 * LDA];
  int t = threadIdx.x;
  int wgRow = blockIdx.x * 64;

  // ---- gather A tile: 64 rows x 96, segment routed by pred, slot-offset ----
  {
    int r = t >> 2;            // local row 0..63
    int j = t & 3;             // 24-float chunk
    int grow = wgRow + r;
    int blk  = grow >> 2;
    int slot = grow & 3;
    int pr = pred[blk];
    int off;
    if (pr == 0)      off = slot * 8;
    else if (pr == 1) off = (slot == 3) ? 16 : 0;
    else              off = 0;
    int qbase = j * 24;
#pragma unroll
    for (int k = 0; k < 24; ++k) {
      int q = qbase + k;
      int seg = q >> 5, qm = q & 31;
      float v = 0.0f;
      if (seg == pr) {
        int xi = blk * 32 + off + qm;   // junk beyond pl hits zero weights
        if (xi >= XLEN) xi = XLEN - 1;  // clamp tail (zero-weighted anyway)
        v = x[xi];
      }
      sA[r * LDA + q] = v;
    }
  }

  int lane  = t & 31;
  int wave  = t >> 5;
  int lhalf = lane & 15;
  int koff  = (lane < 16) ? 0 : 2;   // 16x4 f32 fragment: VGPR0=K0|K2, VGPR1=K1|K3

  __syncthreads();

  for (int cti = 0; cti < 4; ++cti) {
    int ct = wave + cti * 8;
    int colbase = ct * 16;
    // B fragments for this 16-col tile, all 24 K-steps, kept in VGPRs
    v2f bfrag[24];
#pragma unroll
    for (int k = 0; k < 24; ++k) {
      const float* bp = wsB + (size_t)(4 * k + koff) * DM + colbase + lhalf;
      bfrag[k].x = bp[0];
      bfrag[k].y = bp[DM];
    }
    for (int rt = 0; rt < 4; ++rt) {
      int rbase = rt * 16;
      v8f acc = {0.f, 0.f, 0.f, 0.f, 0.f, 0.f, 0.f, 0.f};
      const float* ap = &sA[(rbase + lhalf) * LDA + koff];
#pragma unroll
      for (int k = 0; k < 24; ++k) {
        v2f af;
        af.x = ap[4 * k];
        af.y = ap[4 * k + 1];
        acc = __builtin_amdgcn_wmma_f32_16x16x4_f32(
            false, af, false, bfrag[k], (short)0, acc, false, false);
      }
      // D layout: VGPR v -> M = v (lanes 0-15) / v+8 (lanes 16-31), N = lane%16
      int rowlo = wgRow + rbase + ((lane < 16) ? 0 : 8);
      int col = colbase + lhalf;
#pragma unroll
      for (int v = 0; v < 8; ++v) {
        int grow = rowlo + v;
        out[(size_t)grow * DM + col] = acc[v] + wsPE[(grow & 127) * DM + col];
      }
    }
  }
}

// ---------------------------------------------------------------------------
extern "C" void kernel_launch(void* const* d_in, const int* in_sizes, int n_in,
                              void* d_out, int out_size, void* d_ws, size_t ws_size,
                              hipStream_t stream) {
  const float* x      = (const float*)d_in[0];
  const float* w1     = (const float*)d_in[1];
  const float* b1     = (const float*)d_in[2];
  const float* w2     = (const float*)d_in[3];
  const float* b2     = (const float*)d_in[4];
  const float* w_emb0 = (const float*)d_in[5];
  const float* w_emb1 = (const float*)d_in[6];
  const float* w_emb2 = (const float*)d_in[7];
  float* out = (float*)d_out;

  float* wsf  = (float*)d_ws;
  float* wsB  = wsf;                       // 96*512 floats
  float* wsPE = wsf + 96 * DM;             // 128*512 floats
  int*   pred = (int*)(wsf + 96 * DM + 128 * DM);  // 43008 ints

  int initWork = 96 * DM + 128 * DM;
  init_kernel<<<(initWork + 255) / 256, 256, 0, stream>>>(
      w_emb0, w_emb1, w_emb2, wsB, wsPE, out, out_size);
  classify_kernel<<<(NBLK + 255) / 256, 256, 0, stream>>>(
      x, w1, b1, w2, b2, pred);
  embed_gemm_kernel<<<NROW / 64, 256, 0, stream>>>(
      x, wsB, wsPE, pred, out);
}